// Model_81097572483751
// MI455X (gfx1250) — compile-verified
//
#include <hip/hip_runtime.h>
#include <math.h>

#define BATCH 512
#define NNODE 64
#define NEDGE 1024
#define DIM   128
#define NSTEP 5

typedef __attribute__((ext_vector_type(2))) float v2f;
typedef __attribute__((ext_vector_type(8))) float v8f;

// D = A(16x4 f32) x B(4x16 f32) + C(16x16 f32), wave32 WMMA.
__device__ __forceinline__ v8f wmma4(v2f a, v2f b, v8f c) {
    return __builtin_amdgcn_wmma_f32_16x16x4_f32(
        /*neg_a=*/false, a, /*neg_b=*/false, b,
        /*c_mod=*/(short)0, c, /*reuse_a=*/false, /*reuse_b=*/false);
}

__device__ __forceinline__ float sigmoidf_(float x) {
    return 1.0f / (1.0f + __expf(-x));
}

// ---------------------------------------------------------------------------
// K0: P[t] = transpose((W_link^T)^(t+1)), C[t] = bias accumulation c_{t+1}.
// Stored transposed: P[t][n*128+k] = ((W^T)^(t+1))[k][n] so B-fragments are
// contiguous v2f loads in the consumer kernels.  P[0] == W_link itself.
// ---------------------------------------------------------------------------
__global__ void prep_kernel(const float* __restrict__ Wl, const float* __restrict__ bl,
                            float* __restrict__ P, float* __restrict__ C) {
    __shared__ float cbuf[2][DIM];
    int tid = threadIdx.x;
    for (int idx = tid; idx < DIM * DIM; idx += 256) P[idx] = Wl[idx];
    if (tid < DIM) { cbuf[0][tid] = bl[tid]; C[tid] = bl[tid]; }
    __threadfence();
    __syncthreads();
    for (int t = 1; t < NSTEP; ++t) {
        const float* prev = P + (t - 1) * DIM * DIM;
        for (int idx = tid; idx < DIM * DIM; idx += 256) {
            int n = idx >> 7, k = idx & 127;
            const float* wrow = Wl + n * DIM;
            float acc = 0.0f;
            for (int j = 0; j < DIM; ++j) acc += prev[j * DIM + k] * wrow[j];
            P[t * DIM * DIM + idx] = acc;
        }
        if (tid < DIM) {
            const float* wrow = Wl + tid * DIM;
            float acc = bl[tid];
            for (int j = 0; j < DIM; ++j) acc += cbuf[(t - 1) & 1][j] * wrow[j];
            cbuf[t & 1][tid] = acc;
            C[t * DIM + tid] = acc;
        }
        __threadfence();
        __syncthreads();
    }
}

// ---------------------------------------------------------------------------
// K1: S[b] = A[b](64x1024) * E0[b](1024x128), plus rowsum[b][n] = sum_e A.
// One block per batch, 8 waves; wave w owns output columns [16w,16w+16).
// K staged through LDS in 64-wide chunks.  Deterministic rowsums.
// ---------------------------------------------------------------------------
__global__ void bmm_s_kernel(const float* __restrict__ A, const float* __restrict__ E0,
                             float* __restrict__ S, float* __restrict__ rowsum) {
    __shared__ float lA[64 * 64];    // 16 KB
    __shared__ float lE[64 * 128];   // 32 KB
    __shared__ float pr[256];
    int b    = blockIdx.x;
    int tid  = threadIdx.x;
    int lane = tid & 31;
    int wave = tid >> 5;
    const float* Ab = A  + (size_t)b * NNODE * NEDGE;
    const float* Eb = E0 + (size_t)b * NEDGE * DIM;

    int arow = tid >> 2;            // 0..63 (fixed per thread across chunks)
    int akk0 = (tid & 3) * 16;
    int ekk  = tid >> 2;            // 0..63
    int ed0  = (tid & 3) * 32;

    int n0   = wave * 16;
    int nl   = lane & 15;
    int bn   = n0 + nl;
    int koff = 2 * (lane >> 4);

    float rs = 0.0f;
    v8f acc0 = {}, acc1 = {}, acc2 = {}, acc3 = {};

    for (int kc = 0; kc < 16; ++kc) {
        int k0 = kc * 64;
        {   // stage A chunk [64 x 64], accumulate private rowsum partial
            const float* src = Ab + (size_t)arow * NEDGE + k0 + akk0;
            float* dst = lA + arow * 64 + akk0;
#pragma unroll
            for (int j = 0; j < 16; j += 4) {
                float4 v = *(const float4*)(src + j);
                *(float4*)(dst + j) = v;
                rs += v.x + v.y + v.z + v.w;
            }
            if (kc + 1 < 16) __builtin_prefetch(src + 64, 0, 3);
        }
        {   // stage E chunk [64 x 128]
            const float* src = Eb + (size_t)(k0 + ekk) * DIM + ed0;
            float* dst = lE + ekk * DIM + ed0;
#pragma unroll
            for (int j = 0; j < 32; j += 4)
                *(float4*)(dst + j) = *(const float4*)(src + j);
            if (kc + 1 < 16) __builtin_prefetch(src + 64 * DIM, 0, 3);
        }
        __syncthreads();
        for (int kk = 0; kk < 64; kk += 4) {
            int kb = kk + koff;
            v2f bf;
            bf.x = lE[kb * DIM + bn];
            bf.y = lE[(kb + 1) * DIM + bn];
            v2f a0 = *(const v2f*)&lA[(nl     ) * 64 + kb];
            v2f a1 = *(const v2f*)&lA[(nl + 16) * 64 + kb];
            v2f a2 = *(const v2f*)&lA[(nl + 32) * 64 + kb];
            v2f a3 = *(const v2f*)&lA[(nl + 48) * 64 + kb];
            acc0 = wmma4(a0, bf, acc0);
            acc1 = wmma4(a1, bf, acc1);
            acc2 = wmma4(a2, bf, acc2);
            acc3 = wmma4(a3, bf, acc3);
        }
        __syncthreads();
    }

    int col   = n0 + nl;
    int rbase = 8 * (lane >> 4);
    float* Sb = S + (size_t)b * NNODE * DIM;
#pragma unroll
    for (int v = 0; v < 8; ++v) {
        int r = v + rbase;
        Sb[(r     ) * DIM + col] = acc0[v];
        Sb[(r + 16) * DIM + col] = acc1[v];
        Sb[(r + 32) * DIM + col] = acc2[v];
        Sb[(r + 48) * DIM + col] = acc3[v];
    }
    pr[tid] = rs;
    __syncthreads();
    if (tid < 64)
        rowsum[b * NNODE + tid] = pr[4 * tid] + pr[4 * tid + 1] + pr[4 * tid + 2] + pr[4 * tid + 3];
}

// ---------------------------------------------------------------------------
// K2: per (b, 16-node tile): 5 GRU steps fully resident in LDS, then fused
// attention MLP (logits) and out = tanh(h * W_out^T + b_out).
// a_cur^(t) = S * P_t + rowsum (x) c_t.
// ---------------------------------------------------------------------------
__global__ void prop_kernel(const float* __restrict__ S, const float* __restrict__ rowsum,
                            const float* __restrict__ P, const float* __restrict__ C,
                            const float* __restrict__ h0,
                            const float* __restrict__ Wr, const float* __restrict__ br,
                            const float* __restrict__ Wz, const float* __restrict__ bz,
                            const float* __restrict__ Wh, const float* __restrict__ bh,
                            const float* __restrict__ Wa1, const float* __restrict__ ba1,
                            const float* __restrict__ Wa2, const float* __restrict__ ba2,
                            const float* __restrict__ Wo, const float* __restrict__ bo,
                            float* __restrict__ out_nodes, float* __restrict__ logits) {
    __shared__ float sS[16 * DIM];   // S tile
    __shared__ float sH[16 * DIM];   // current hidden state
    __shared__ float sA[16 * DIM];   // a_cur (later: att1)
    __shared__ float sR[16 * DIM];   // r * h
    __shared__ float sRS[16];

    int blk  = blockIdx.x;
    int b    = blk >> 2;
    int nb   = (blk & 3) * 16;
    int tid  = threadIdx.x;
    int lane = tid & 31;
    int wave = tid >> 5;

    for (int idx = tid; idx < 16 * DIM; idx += 256) {
        int m = idx >> 7, d = idx & 127;
        size_t g = ((size_t)b * NNODE + nb + m) * DIM + d;
        sS[idx] = S[g];
        sH[idx] = h0[g];
    }
    if (tid < 16) sRS[tid] = rowsum[b * NNODE + nb + tid];
    __syncthreads();

    int n0    = wave * 16;
    int nl    = lane & 15;
    int col   = n0 + nl;
    int koff  = 2 * (lane >> 4);
    int rbase = 8 * (lane >> 4);

    for (int t = 0; t < NSTEP; ++t) {
        const float* Pt = P + t * DIM * DIM;   // transposed power
        const float* Ct = C + t * DIM;
        // ---- 1) a_cur tile ----
        v8f acc = {};
        for (int k = 0; k < DIM; k += 4) {
            int kb = k + koff;
            v2f af = *(const v2f*)&sS[nl * DIM + kb];
            v2f bf = *(const v2f*)&Pt[col * DIM + kb];
            acc = wmma4(af, bf, acc);
        }
        float cn = Ct[col];
#pragma unroll
        for (int v = 0; v < 8; ++v) {
            int r = v + rbase;
            sA[r * DIM + col] = acc[v] + sRS[r] * cn;
        }
        __syncthreads();
        // ---- 2) r and z gates over concat [a_cur, h] (K = 256) ----
        v8f accr = {}, accz = {};
        for (int k = 0; k < DIM; k += 4) {          // first half: a_cur
            int kb = k + koff;
            v2f af = *(const v2f*)&sA[nl * DIM + kb];
            v2f bfr = *(const v2f*)&Wr[col * 2 * DIM + kb];
            v2f bfz = *(const v2f*)&Wz[col * 2 * DIM + kb];
            accr = wmma4(af, bfr, accr);
            accz = wmma4(af, bfz, accz);
        }
        for (int k = 0; k < DIM; k += 4) {          // second half: h
            int kb = k + koff;
            v2f af = *(const v2f*)&sH[nl * DIM + kb];
            v2f bfr = *(const v2f*)&Wr[col * 2 * DIM + DIM + kb];
            v2f bfz = *(const v2f*)&Wz[col * 2 * DIM + DIM + kb];
            accr = wmma4(af, bfr, accr);
            accz = wmma4(af, bfz, accz);
        }
        float bias_r = br[col], bias_z = bz[col];
        v8f zreg;
#pragma unroll
        for (int v = 0; v < 8; ++v) {
            int r = v + rbase;
            float rv = sigmoidf_(accr[v] + bias_r);
            sR[r * DIM + col] = rv * sH[r * DIM + col];
            zreg[v] = sigmoidf_(accz[v] + bias_z);
        }
        __syncthreads();
        // ---- 3) h_hat over concat [a_cur, r*h]; GRU update ----
        v8f acch = {};
        for (int k = 0; k < DIM; k += 4) {
            int kb = k + koff;
            v2f af = *(const v2f*)&sA[nl * DIM + kb];
            v2f bf = *(const v2f*)&Wh[col * 2 * DIM + kb];
            acch = wmma4(af, bf, acch);
        }
        for (int k = 0; k < DIM; k += 4) {
            int kb = k + koff;
            v2f af = *(const v2f*)&sR[nl * DIM + kb];
            v2f bf = *(const v2f*)&Wh[col * 2 * DIM + DIM + kb];
            acch = wmma4(af, bf, acch);
        }
        float bias_h = bh[col];
#pragma unroll
        for (int v = 0; v < 8; ++v) {
            int r = v + rbase;
            float hh   = tanhf(acch[v] + bias_h);
            float hold = sH[r * DIM + col];
            sH[r * DIM + col] = (1.0f - zreg[v]) * hold + zreg[v] * hh;   // own cols only
        }
        __syncthreads();
    }

    // ---- attention MLP: att1 = tanh(h * Wa1^T + ba1) -> sA ----
    v8f acca = {};
    for (int k = 0; k < DIM; k += 4) {
        int kb = k + koff;
        v2f af = *(const v2f*)&sH[nl * DIM + kb];
        v2f bf = *(const v2f*)&Wa1[col * DIM + kb];
        acca = wmma4(af, bf, acca);
    }
    float ba = ba1[col];
#pragma unroll
    for (int v = 0; v < 8; ++v)
        sA[(v + rbase) * DIM + col] = tanhf(acca[v] + ba);
    __syncthreads();
    if (tid < 16) {
        float acc = ba2[0];
        for (int d = 0; d < DIM; ++d) acc += sA[tid * DIM + d] * Wa2[d];
        logits[b * NNODE + nb + tid] = acc;
    }
    // ---- out = tanh(h * Wo^T + bo) ----
    v8f acco = {};
    for (int k = 0; k < DIM; k += 4) {
        int kb = k + koff;
        v2f af = *(const v2f*)&sH[nl * DIM + kb];
        v2f bf = *(const v2f*)&Wo[col * DIM + kb];
        acco = wmma4(af, bf, acco);
    }
    float bb = bo[col];
#pragma unroll
    for (int v = 0; v < 8; ++v) {
        int r = v + rbase;
        out_nodes[((size_t)b * NNODE + nb + r) * DIM + col] = tanhf(acco[v] + bb);
    }
}

// ---------------------------------------------------------------------------
// K4: edge_out = E0 * (W^T)^5 + 1 (x) c5, plus per-block column partial sums
// of E0 for the residual mean (deterministic, no atomics).
// One block per 64 edges; 8 waves, wave w owns cols [16w,16w+16), 4 M-tiles.
// ---------------------------------------------------------------------------
__global__ void edge_kernel(const float* __restrict__ E0, const float* __restrict__ PT5,
                            const float* __restrict__ C5,
                            float* __restrict__ edge_out, float* __restrict__ edge_part) {
    __shared__ float lE[64 * DIM];   // 32 KB
    int blk  = blockIdx.x;
    int b    = blk >> 4;
    int et   = blk & 15;
    int eb   = et * 64;
    int tid  = threadIdx.x;
    int lane = tid & 31;
    int wave = tid >> 5;
    const float* Eb = E0 + ((size_t)b * NEDGE + eb) * DIM;

    int ekk = tid >> 2, ed0 = (tid & 3) * 32;
    {
        const float* src = Eb + (size_t)ekk * DIM + ed0;
        float* dst = lE + ekk * DIM + ed0;
#pragma unroll
        for (int j = 0; j < 32; j += 4)
            *(float4*)(dst + j) = *(const float4*)(src + j);
    }
    __syncthreads();
    if (tid < DIM) {   // column partial sums for residual
        float s = 0.0f;
        for (int r = 0; r < 64; ++r) s += lE[r * DIM + tid];
        edge_part[((size_t)b * 16 + et) * DIM + tid] = s;
    }

    int n0 = wave * 16, nl = lane & 15, col = n0 + nl;
    int koff = 2 * (lane >> 4), rbase = 8 * (lane >> 4);
    v8f acc0 = {}, acc1 = {}, acc2 = {}, acc3 = {};
    for (int k = 0; k < DIM; k += 4) {
        int kb = k + koff;
        v2f bf = *(const v2f*)&PT5[col * DIM + kb];
        v2f a0 = *(const v2f*)&lE[(nl     ) * DIM + kb];
        v2f a1 = *(const v2f*)&lE[(nl + 16) * DIM + kb];
        v2f a2 = *(const v2f*)&lE[(nl + 32) * DIM + kb];
        v2f a3 = *(const v2f*)&lE[(nl + 48) * DIM + kb];
        acc0 = wmma4(a0, bf, acc0);
        acc1 = wmma4(a1, bf, acc1);
        acc2 = wmma4(a2, bf, acc2);
        acc3 = wmma4(a3, bf, acc3);
    }
    float cn = C5[col];
    float* Ob = edge_out + ((size_t)b * NEDGE + eb) * DIM;
#pragma unroll
    for (int v = 0; v < 8; ++v) {
        int r = v + rbase;
        Ob[(r     ) * DIM + col] = acc0[v] + cn;
        Ob[(r + 16) * DIM + col] = acc1[v] + cn;
        Ob[(r + 32) * DIM + col] = acc2[v] + cn;
        Ob[(r + 48) * DIM + col] = acc3[v] + cn;
    }
}

// ---------------------------------------------------------------------------
// K3: softmax attention pool + residual mean + relu -> res[B, D].
// ---------------------------------------------------------------------------
__global__ void final_kernel(const float* __restrict__ init_nodes,
                             const float* __restrict__ logits,
                             const float* __restrict__ out_nodes,
                             const float* __restrict__ edge_part,
                             float* __restrict__ res) {
    __shared__ float satt[NNODE];
    __shared__ float sinv;
    int b = blockIdx.x, tid = threadIdx.x;
    if (tid < NNODE) satt[tid] = logits[b * NNODE + tid];
    __syncthreads();
    if (tid == 0) {
        float m = satt[0];
        for (int n = 1; n < NNODE; ++n) m = fmaxf(m, satt[n]);
        float s = 0.0f;
        for (int n = 0; n < NNODE; ++n) { satt[n] = __expf(satt[n] - m); s += satt[n]; }
        sinv = 1.0f / s;
    }
    __syncthreads();
    int d = tid;   // 128 threads
    float nsum = 0.0f;
    for (int n = 0; n < NNODE; ++n) nsum += init_nodes[((size_t)b * NNODE + n) * DIM + d];
    float esum = 0.0f;
    for (int p = 0; p < 16; ++p) esum += edge_part[((size_t)b * 16 + p) * DIM + d];
    float residual = (nsum + esum) * (1.0f / (float)(NNODE + NEDGE));
    float wsum = 0.0f;
    for (int n = 0; n < NNODE; ++n) wsum += satt[n] * out_nodes[((size_t)b * NNODE + n) * DIM + d];
    float g = tanhf(wsum * sinv);
    res[(size_t)b * DIM + d] = fmaxf(g + residual, 0.0f);
}

// ---------------------------------------------------------------------------
extern "C" void kernel_launch(void* const* d_in, const int* in_sizes, int n_in,
                              void* d_out, int out_size, void* d_ws, size_t ws_size,
                              hipStream_t stream) {
    (void)in_sizes; (void)n_in; (void)out_size; (void)ws_size;
    const float* prop = (const float*)d_in[0];
    const float* edges = (const float*)d_in[1];
    const float* A   = (const float*)d_in[2];
    const float* Wl  = (const float*)d_in[3];
    const float* bl  = (const float*)d_in[4];
    const float* Wr  = (const float*)d_in[5];
    const float* br  = (const float*)d_in[6];
    const float* Wz  = (const float*)d_in[7];
    const float* bz  = (const float*)d_in[8];
    const float* Wh  = (const float*)d_in[9];
    const float* bh  = (const float*)d_in[10];
    const float* Wa1 = (const float*)d_in[11];
    const float* ba1 = (const float*)d_in[12];
    const float* Wa2 = (const float*)d_in[13];
    const float* ba2 = (const float*)d_in[14];
    const float* Wo  = (const float*)d_in[15];
    const float* bo  = (const float*)d_in[16];

    float* ws        = (float*)d_ws;
    float* P         = ws;                                   // 5*128*128 (transposed powers)
    float* C         = P + NSTEP * DIM * DIM;                // 5*128
    float* S         = C + NSTEP * DIM;                      // 512*64*128
    float* rowsum    = S + (size_t)BATCH * NNODE * DIM;      // 512*64
    float* logits    = rowsum + BATCH * NNODE;               // 512*64
    float* edge_part = logits + BATCH * NNODE;               // 512*16*128

    float* res       = (float*)d_out;                        // 512*128
    float* out_nodes = res + (size_t)BATCH * DIM;            // 512*64*128
    float* edge_out  = out_nodes + (size_t)BATCH * NNODE * DIM;  // 512*1024*128

    prep_kernel <<<1, 256, 0, stream>>>(Wl, bl, P, C);
    bmm_s_kernel<<<BATCH, 256, 0, stream>>>(A, edges, S, rowsum);
    prop_kernel <<<BATCH * 4, 256, 0, stream>>>(S, rowsum, P, C, prop,
                                                Wr, br, Wz, bz, Wh, bh,
                                                Wa1, ba1, Wa2, ba2, Wo, bo,
                                                out_nodes, logits);
    edge_kernel <<<BATCH * 16, 256, 0, stream>>>(edges, P + 4 * DIM * DIM, C + 4 * DIM,
                                                 edge_out, edge_part);
    final_kernel<<<BATCH, 128, 0, stream>>>(prop, logits, out_nodes, edge_part, res);
}